// PartialCrossEntropyLoss_15333033247373
// MI455X (gfx1250) — compile-verified
//
#include <hip/hip_runtime.h>
#include <hip/hip_bf16.h>

// PartialCrossEntropyLoss for MI455X (gfx1250, wave32).
// y_pred (8,1,512,512) f32, point_labels (8,20) i32, point_coords (8,20,2) f32.
// Output: scalar f32 loss.

#define BB 8
#define NP 20
#define HH 512
#define WW 512
#define HWPIX (HH * WW)          // 262144
#define PIX_PER_BLK 2048         // divides HWPIX -> b constant per block
#define NBLK ((BB * HWPIX) / PIX_PER_BLK)   // 1024

// exp(-d / (2*sigma^2)) > 0.1  <=>  d < 50*ln(10)
#define DIST_T 115.12925464970229f

typedef __attribute__((ext_vector_type(2))) float v2f;
typedef __attribute__((ext_vector_type(8))) float v8f;

// Wave-wide reduction of (sumB, sumC) pairs using V_WMMA_F32_16X16X4_F32.
// A (16x4): lane L<16 holds A[L,0]=sumB_L, A[L,1]=sumC_L;
//           lane 16+m holds A[m,2]=sumB_{16+m}, A[m,3]=sumC_{16+m}.
// Selector B: B[0,0]=B[2,0]=1 (col 0 = bce), B[1,1]=B[3,1]=1 (col 1 = cnt).
// After summing the 8 D VGPRs per lane and one shfl_down(16):
//   lane 0 -> wave bce total, lane 1 -> wave cnt total.
__device__ inline float wave_pair_reduce_wmma(float sumB, float sumC, int lane) {
    v2f a;  a.x = sumB;  a.y = sumC;
    v2f bs; bs.x = ((lane & 15) == 0) ? 1.0f : 0.0f;
            bs.y = ((lane & 15) == 1) ? 1.0f : 0.0f;
    v8f c = {};
    v8f d = __builtin_amdgcn_wmma_f32_16x16x4_f32(
        false, a, false, bs, (short)0, c, false, false);
    float s = d[0] + d[1] + d[2] + d[3] + d[4] + d[5] + d[6] + d[7];
    s += __shfl_down(s, 16, 32);
    return s;
}

__global__ __launch_bounds__(256)
void pce_main_kernel(const float* __restrict__ y_pred,
                     const int* __restrict__ point_labels,
                     const float* __restrict__ point_coords,
                     float* __restrict__ partials) {
    __shared__ float4 s_pt[NP];          // {yc, xc, label, 0}
    __shared__ float  s_wb[8], s_wc[8];

    const int t    = threadIdx.x;
    const int blk  = blockIdx.x;
    const int base = blk * PIX_PER_BLK;
    const int b    = base / HWPIX;       // uniform per block

    if (t < NP) {
        const int pi = b * NP + t;
        float4 pt;
        pt.x = point_coords[pi * 2 + 0] * (float)HH;   // yc
        pt.y = point_coords[pi * 2 + 1] * (float)WW;   // xc
        pt.z = (float)point_labels[pi];                // label
        pt.w = 0.0f;
        s_pt[t] = pt;
    }
    __syncthreads();

    // Prefetch the second half-chunk (emits global_prefetch_b8).
    __builtin_prefetch(y_pred + base + 1024 + t * 4, 0, 0);

    float sumB = 0.0f, sumC = 0.0f;

#pragma unroll
    for (int r = 0; r < 2; ++r) {
        const int p = base + r * 1024 + t * 4;   // 4 contiguous pixels, same row
        const float4 v = *reinterpret_cast<const float4*>(y_pred + p);
        const int q  = p - b * HWPIX;
        const float yf = (float)(q >> 9);
        const float xf = (float)(q & 511);
        const float x0 = xf, x1 = xf + 1.0f, x2 = xf + 2.0f, x3 = xf + 3.0f;

        // Sentinel -1: "no point hit". Labels are 0/1.
        float lb0 = -1.0f, lb1 = -1.0f, lb2 = -1.0f, lb3 = -1.0f;

        // Ascending n; select overrides => label of the max-n ("last") hit.
        // Branchless inner body; only the wave-coherent row cull branches.
        for (int n = 0; n < NP; ++n) {
            const float4 pt = s_pt[n];
            const float dy  = yf - pt.x;
            const float dy2 = dy * dy;
            if (dy2 < DIST_T) {
                const float xc = pt.y, lab = pt.z;
                float dx, d;
                dx = x0 - xc; d = fmaf(dx, dx, dy2); lb0 = (d < DIST_T) ? lab : lb0;
                dx = x1 - xc; d = fmaf(dx, dx, dy2); lb1 = (d < DIST_T) ? lab : lb1;
                dx = x2 - xc; d = fmaf(dx, dx, dy2); lb2 = (d < DIST_T) ? lab : lb2;
                dx = x3 - xc; d = fmaf(dx, dx, dy2); lb3 = (d < DIST_T) ? lab : lb3;
            }
        }

        const float lg[4] = { v.x, v.y, v.z, v.w };
        const float lb[4] = { lb0, lb1, lb2, lb3 };
#pragma unroll
        for (int j = 0; j < 4; ++j) {
            if (lb[j] > -0.5f) {     // valid: some point hit this pixel
                const float l   = lg[j];
                const float bce = fmaxf(l, 0.0f) - l * lb[j]
                                + log1pf(expf(-fabsf(l)));
                sumB += bce;
                sumC += 1.0f;
            }
        }
    }

    // Wave-level reduce via WMMA (full EXEC here: straight-line code).
    const int lane = t & 31;
    const int wv   = t >> 5;
    const float s  = wave_pair_reduce_wmma(sumB, sumC, lane);
    if (lane == 0) s_wb[wv] = s;
    if (lane == 1) s_wc[wv] = s;
    __syncthreads();

    if (t == 0) {
        float tb = 0.0f, tc = 0.0f;
#pragma unroll
        for (int i = 0; i < 8; ++i) { tb += s_wb[i]; tc += s_wc[i]; }
        partials[2 * blk + 0] = tb;
        partials[2 * blk + 1] = tc;
    }
}

__global__ __launch_bounds__(32)
void pce_final_kernel(const float* __restrict__ partials,
                      float* __restrict__ out) {
    const int lane = threadIdx.x;    // one wave32
    float sb = 0.0f, sc = 0.0f;
    // Fixed-order deterministic accumulation: 1024/32 = 32 iters per lane.
    for (int k = lane; k < NBLK; k += 32) {
        sb += partials[2 * k + 0];
        sc += partials[2 * k + 1];
    }
    const float s   = wave_pair_reduce_wmma(sb, sc, lane);
    const float cnt = __shfl(s, 1, 32);   // broadcast lane 1's count
    if (lane == 0) {
        out[0] = (cnt > 0.0f) ? (s / cnt) : 0.0f;
    }
}

extern "C" void kernel_launch(void* const* d_in, const int* in_sizes, int n_in,
                              void* d_out, int out_size, void* d_ws, size_t ws_size,
                              hipStream_t stream) {
    (void)in_sizes; (void)n_in; (void)out_size; (void)ws_size;
    const float* y_pred       = (const float*)d_in[0];
    const int*   point_labels = (const int*)d_in[1];
    const float* point_coords = (const float*)d_in[2];
    float*       out          = (float*)d_out;
    float*       partials     = (float*)d_ws;   // NBLK * 2 floats = 8 KiB

    pce_main_kernel<<<NBLK, 256, 0, stream>>>(y_pred, point_labels, point_coords, partials);
    pce_final_kernel<<<1, 32, 0, stream>>>(partials, out);
}